// ConvolutionBlock_61040075211152
// MI455X (gfx1250) — compile-verified
//
#include <hip/hip_runtime.h>
#include <math.h>

// ---------------- problem constants ----------------
#define B_   16
#define CIN  64
#define T0   4096
#define CO   32
#define T1   2048          // after GLU over time
#define T2   2019          // after depthwise conv k=32 pad=1
#define H_   64
#define NS   64            // S4D state size
#define NL   8
#define EPSN 1e-5f

typedef __attribute__((ext_vector_type(2))) float v2f;
typedef __attribute__((ext_vector_type(8))) float v8f;

// =====================================================================
// Generic f32 WMMA GEMM with 1x4 tile blocking per wave:
// each wave computes a 16x64 strip of C[b] = A[b](MxK)*B[b](KxN) (+bias).
// The 16x4 A fragment is loaded once per K-step and reused by 4
// V_WMMA_F32_16X16X4_F32 issues (operand reuse, amortized addressing).
// A layout (16x4 f32): lanes 0-15 hold K=k,k+1 ; lanes 16-31 K=k+2,k+3.
// B layout (4x16 f32): mirrored. C/D: VGPR i -> row m0+i (lanes 0-15) /
// m0+i+8 (lanes 16-31), col = n + (lane&15).
// =====================================================================
__global__ __launch_bounds__(32)
void wmma_gemm_f32(const float* __restrict__ A, long long a_bs, int a_rs, int a_cs,
                   const float* __restrict__ Bm, long long b_bs, int b_rs, int b_cs,
                   float* __restrict__ C, long long c_bs, int c_rs, int c_cs,
                   const float* __restrict__ bias, int bias_mode,
                   int M, int N, int K)
{
    const int lane = threadIdx.x & 31;
    const int n0 = blockIdx.x * 64;
    const int m0 = blockIdx.y * 16;
    A  += (long long)blockIdx.z * a_bs;
    Bm += (long long)blockIdx.z * b_bs;
    C  += (long long)blockIdx.z * c_bs;

    const int hi  = lane >> 4;      // which K pair this half-wave holds
    const int l15 = lane & 15;
    int am = m0 + l15; if (am >= M) am = M - 1;       // clamp; masked at store
    int bn[4];
#pragma unroll
    for (int s = 0; s < 4; ++s) {
        int c = n0 + s * 16 + l15;
        bn[s] = (c < N) ? c : (N - 1);
    }

    const float* __restrict__ Ap = A + (long long)am * a_rs;

    v8f acc0 = {0.f,0.f,0.f,0.f,0.f,0.f,0.f,0.f};
    v8f acc1 = acc0, acc2 = acc0, acc3 = acc0;

    for (int k = 0; k < K; k += 4) {
        const int ka = k + hi * 2;
        v2f a;
        a.x = Ap[(long long)ka * a_cs];
        a.y = Ap[(long long)(ka + 1) * a_cs];
        const float* __restrict__ Br0 = Bm + (long long)ka * b_rs;
        const float* __restrict__ Br1 = Bm + (long long)(ka + 1) * b_rs;
        v2f b0, b1, b2, b3;
        b0.x = Br0[(long long)bn[0] * b_cs];  b0.y = Br1[(long long)bn[0] * b_cs];
        b1.x = Br0[(long long)bn[1] * b_cs];  b1.y = Br1[(long long)bn[1] * b_cs];
        b2.x = Br0[(long long)bn[2] * b_cs];  b2.y = Br1[(long long)bn[2] * b_cs];
        b3.x = Br0[(long long)bn[3] * b_cs];  b3.y = Br1[(long long)bn[3] * b_cs];
        acc0 = __builtin_amdgcn_wmma_f32_16x16x4_f32(false, a, false, b0, (short)0, acc0, false, false);
        acc1 = __builtin_amdgcn_wmma_f32_16x16x4_f32(false, a, false, b1, (short)0, acc1, false, false);
        acc2 = __builtin_amdgcn_wmma_f32_16x16x4_f32(false, a, false, b2, (short)0, acc2, false, false);
        acc3 = __builtin_amdgcn_wmma_f32_16x16x4_f32(false, a, false, b3, (short)0, acc3, false, false);
    }

    v8f accs[4] = {acc0, acc1, acc2, acc3};
#pragma unroll
    for (int s = 0; s < 4; ++s) {
        const int col = n0 + s * 16 + l15;
        if (col < N) {
#pragma unroll
            for (int i = 0; i < 8; ++i) {
                const int row = m0 + i + hi * 8;
                if (row < M) {
                    float v = accs[s][i];
                    if (bias_mode == 1)      v += bias[row];
                    else if (bias_mode == 2) v += bias[col];
                    C[(long long)row * c_rs + (long long)col * c_cs] = v;
                }
            }
        }
    }
}

// =====================================================================
// GroupNorm(1,C) stats over (C,T) per example
// =====================================================================
__global__ void k_gn_stats(const float* __restrict__ u, float* __restrict__ mean,
                           float* __restrict__ rstd, int CT)
{
    __shared__ float ssum[256], ssq[256];
    const int b = blockIdx.x;
    const float* p = u + (long long)b * CT;
    float s = 0.f, q = 0.f;
    for (int i = threadIdx.x; i < CT; i += 256) { float v = p[i]; s += v; q += v * v; }
    ssum[threadIdx.x] = s; ssq[threadIdx.x] = q; __syncthreads();
    for (int off = 128; off > 0; off >>= 1) {
        if (threadIdx.x < off) {
            ssum[threadIdx.x] += ssum[threadIdx.x + off];
            ssq[threadIdx.x]  += ssq[threadIdx.x + off];
        }
        __syncthreads();
    }
    if (threadIdx.x == 0) {
        float m = ssum[0] / (float)CT;
        float v = ssq[0] / (float)CT - m * m;
        mean[b] = m; rstd[b] = rsqrtf(v + EPSN);
    }
}

// gn1 apply + GLU over time + SiLU : (B,CO,T0) -> (B,CO,T1)
__global__ void k_gn1_glu_silu(const float* __restrict__ u1, const float* __restrict__ mean,
                               const float* __restrict__ rstd, const float* __restrict__ g,
                               const float* __restrict__ bt, float* __restrict__ u2)
{
    int i = blockIdx.x * 256 + threadIdx.x;
    if (i >= B_ * CO * T1) return;
    int t = i % T1, c = (i / T1) % CO, b = i / (T1 * CO);
    const float* p = u1 + ((long long)b * CO + c) * T0;
    float m = mean[b], r = rstd[b];
    float ga = (p[t]      - m) * r * g[c] + bt[c];
    float gb = (p[t + T1] - m) * r * g[c] + bt[c];
    float v = ga * (1.f / (1.f + expf(-gb)));   // GLU
    v = v * (1.f / (1.f + expf(-v)));           // SiLU
    u2[i] = v;
}

// depthwise conv, kernel 32, pad 1 : (B,CO,T1) -> (B,CO,T2)
__global__ void k_dwconv(const float* __restrict__ u2, const float* __restrict__ w,
                         const float* __restrict__ bias, float* __restrict__ u3)
{
    int t = blockIdx.x * 256 + threadIdx.x;
    int c = blockIdx.y, b = blockIdx.z;
    if (t >= T2) return;
    const float* up = u2 + ((long long)b * CO + c) * T1;
    const float* wp = w + c * 32;
    float acc = bias[c];
#pragma unroll
    for (int k = 0; k < 32; ++k) {
        int idx = t + k - 1;
        float v = (idx >= 0 && idx < T1) ? up[idx] : 0.f;
        acc = fmaf(wp[k], v, acc);
    }
    u3[((long long)b * CO + c) * T2 + t] = acc;
}

// gn2 apply (elementwise)
__global__ void k_gn2_apply(const float* __restrict__ u3, const float* __restrict__ mean,
                            const float* __restrict__ rstd, const float* __restrict__ g,
                            const float* __restrict__ bt, float* __restrict__ u4)
{
    int i = blockIdx.x * 256 + threadIdx.x;
    if (i >= B_ * CO * T2) return;
    int c = (i / T2) % CO, b = i / (T2 * CO);
    u4[i] = (u3[i] - mean[b]) * rstd[b] * g[c] + bt[c];
}

// =====================================================================
// S4D discretization tables: w = exp(dt*A), cd = 2*(Cre+iCim)*(w-1)/A
// over all layers at once: (NL,H,NS)
// =====================================================================
__global__ void k_tables(const float* __restrict__ log_dt, const float* __restrict__ logA,
                         const float* __restrict__ Aim, const float* __restrict__ Cre,
                         const float* __restrict__ Cim,
                         float* __restrict__ wre, float* __restrict__ wim,
                         float* __restrict__ cre, float* __restrict__ cim)
{
    int i = blockIdx.x * 256 + threadIdx.x;
    if (i >= NL * H_ * NS) return;
    int hn = i % (H_ * NS);
    int l  = i / (H_ * NS);
    int h  = hn / NS;
    float dt  = expf(log_dt[l * H_ + h]);
    float Are = -expf(logA[i]);
    float Ai  = Aim[i];
    float er  = expf(Are * dt);
    float ere = er * cosf(Ai * dt);
    float eim = er * sinf(Ai * dt);
    float den = Are * Are + Ai * Ai;
    float nre = ere - 1.f, nim = eim;
    float qre = (nre * Are + nim * Ai) / den;   // (w-1)/A
    float qim = (nim * Are - nre * Ai) / den;
    float cr = Cre[i], ci = Cim[i];
    wre[i] = ere;  wim[i] = eim;
    cre[i] = 2.f * (cr * qre - ci * qim);
    cim[i] = 2.f * (cr * qim + ci * qre);
}

// =====================================================================
// S4D scan: one wave32 per (b,h); lane owns states n and n+32.
// s_t = w*s_{t-1} + u_t ; y_t = Re(sum cd*s) + D*u_t ; exact GELU.
// n-reduction = 5 shfl_xor steps (pure intra-wave, wave32-native).
// Coalesced y stores: lane j latches y at t%32==j, flush every 32 steps.
// =====================================================================
__global__ __launch_bounds__(32)
void k_scan(const float* __restrict__ hbuf, const float* __restrict__ wre,
            const float* __restrict__ wim, const float* __restrict__ cre,
            const float* __restrict__ cim, const float* __restrict__ Dp,
            float* __restrict__ y)
{
    const int lane = threadIdx.x;
    const int hh = blockIdx.x % H_;
    const int b  = blockIdx.x / H_;
    const float* up = hbuf + ((long long)b * H_ + hh) * T2;
    float* yp = y + ((long long)b * H_ + hh) * T2;
    const int bn = hh * NS;
    const float w0r = wre[bn + lane],      w0i = wim[bn + lane];
    const float w1r = wre[bn + lane + 32], w1i = wim[bn + lane + 32];
    const float c0r = cre[bn + lane],      c0i = cim[bn + lane];
    const float c1r = cre[bn + lane + 32], c1i = cim[bn + lane + 32];
    const float D = Dp[hh];
    float s0r = 0.f, s0i = 0.f, s1r = 0.f, s1i = 0.f;
    for (int base = 0; base < T2; base += 32) {
        int lim = T2 - base; if (lim > 32) lim = 32;
        float uv = (lane < lim) ? up[base + lane] : 0.f;
        float yreg = 0.f;
        for (int j = 0; j < lim; ++j) {
            float ut = __shfl(uv, j, 32);
            float t0r = w0r * s0r - w0i * s0i + ut;
            float t0i = w0r * s0i + w0i * s0r;
            s0r = t0r; s0i = t0i;
            float t1r = w1r * s1r - w1i * s1i + ut;
            float t1i = w1r * s1i + w1i * s1r;
            s1r = t1r; s1i = t1i;
            float p = c0r * s0r - c0i * s0i + c1r * s1r - c1i * s1i;
            p += __shfl_xor(p, 16, 32);
            p += __shfl_xor(p, 8, 32);
            p += __shfl_xor(p, 4, 32);
            p += __shfl_xor(p, 2, 32);
            p += __shfl_xor(p, 1, 32);
            float yv = p + D * ut;
            yv = 0.5f * yv * (1.f + erff(yv * 0.70710678118f));  // exact GELU
            if (j == lane) yreg = yv;
        }
        if (lane < lim) yp[base + lane] = yreg;
    }
}

// channel-GLU + residual + channel LayerNorm, h updated in place.
// One thread per (b,t): reads of fixed d across adjacent t coalesce.
__global__ void k_glu_res_ln(const float* __restrict__ o, float* __restrict__ h,
                             const float* __restrict__ g, const float* __restrict__ bt)
{
    int i = blockIdx.x * 256 + threadIdx.x;
    if (i >= B_ * T2) return;
    int t = i % T2, b = i / T2;
    const float* ob = o + (long long)b * (2 * H_) * T2 + t;
    float* hb = h + (long long)b * H_ * T2 + t;
    float s = 0.f, q = 0.f;
    for (int d = 0; d < H_; ++d) {
        float oa = ob[(long long)d * T2];
        float og = ob[(long long)(d + H_) * T2];
        float z = oa / (1.f + expf(-og));
        float r = z + hb[(long long)d * T2];
        s += r; q += r * r;
    }
    float m = s / (float)H_;
    float var = q / (float)H_ - m * m;
    float rs = rsqrtf(var + EPSN);
    for (int d = 0; d < H_; ++d) {
        float oa = ob[(long long)d * T2];
        float og = ob[(long long)(d + H_) * T2];
        float z = oa / (1.f + expf(-og));
        float r = z + hb[(long long)d * T2];
        hb[(long long)d * T2] = (r - m) * rs * g[d] + bt[d];
    }
}

// =====================================================================
extern "C" void kernel_launch(void* const* d_in, const int* in_sizes, int n_in,
                              void* d_out, int out_size, void* d_ws, size_t ws_size,
                              hipStream_t stream)
{
    const float* x      = (const float*)d_in[0];
    const float* pw1_w  = (const float*)d_in[1];
    const float* pw1_b  = (const float*)d_in[2];
    const float* gn1_g  = (const float*)d_in[3];
    const float* gn1_b  = (const float*)d_in[4];
    const float* dw_w   = (const float*)d_in[5];
    const float* dw_b   = (const float*)d_in[6];
    const float* gn2_g  = (const float*)d_in[7];
    const float* gn2_b  = (const float*)d_in[8];
    const float* pw2_w  = (const float*)d_in[9];
    const float* pw2_b  = (const float*)d_in[10];
    const float* enc_w  = (const float*)d_in[11];
    const float* enc_b  = (const float*)d_in[12];
    const float* log_dt = (const float*)d_in[13];
    const float* logA   = (const float*)d_in[14];
    const float* Aim    = (const float*)d_in[15];
    const float* Cre    = (const float*)d_in[16];
    const float* Cim    = (const float*)d_in[17];
    const float* s4_D   = (const float*)d_in[18];
    const float* s4_ow  = (const float*)d_in[19];
    const float* s4_ob  = (const float*)d_in[20];
    const float* ln_g   = (const float*)d_in[21];
    const float* ln_b   = (const float*)d_in[22];
    const float* dec_w  = (const float*)d_in[23];
    const float* dec_b  = (const float*)d_in[24];
    float* out = (float*)d_out;

    // ---- workspace layout (floats) ----
    float* W = (float*)d_ws;
    float* u1 = W;                                // B*CO*T0 = 2,097,152
    float* u2 = u1 + (long long)B_ * CO * T0;     // B*CO*T1 = 1,048,576
    float* u3 = u2 + (long long)B_ * CO * T1;     // B*CO*T2 = 1,033,728
    float* u4 = u3 + (long long)B_ * CO * T2;
    float* u5 = u4 + (long long)B_ * CO * T2;
    float* hb = u5 + (long long)B_ * CO * T2;     // B*H*T2 = 2,067,456
    float* yb = hb + (long long)B_ * H_ * T2;
    float* ob = yb + (long long)B_ * H_ * T2;     // B*128*T2 = 4,134,912
    float* st = ob + (long long)B_ * 2 * H_ * T2; // 64 stats
    float* mean1 = st, *rstd1 = st + 16, *mean2 = st + 32, *rstd2 = st + 48;
    float* wre = st + 64;                         // NL*H*NS each
    float* wim = wre + NL * H_ * NS;
    float* cre = wim + NL * H_ * NS;
    float* cim = cre + NL * H_ * NS;

    // S4D discretization tables (all layers)
    k_tables<<<(NL * H_ * NS + 255) / 256, 256, 0, stream>>>(
        log_dt, logA, Aim, Cre, Cim, wre, wim, cre, cim);

    // pw1: u1[b] = pw1_w(32x64) * x[b](64x4096) + b
    {
        dim3 g((T0 + 63) / 64, CO / 16, B_);
        wmma_gemm_f32<<<g, 32, 0, stream>>>(pw1_w, 0, CIN, 1,
                                            x, (long long)CIN * T0, T0, 1,
                                            u1, (long long)CO * T0, T0, 1,
                                            pw1_b, 1, CO, T0, CIN);
    }
    k_gn_stats<<<B_, 256, 0, stream>>>(u1, mean1, rstd1, CO * T0);
    k_gn1_glu_silu<<<(B_ * CO * T1 + 255) / 256, 256, 0, stream>>>(
        u1, mean1, rstd1, gn1_g, gn1_b, u2);
    {
        dim3 g((T2 + 255) / 256, CO, B_);
        k_dwconv<<<g, 256, 0, stream>>>(u2, dw_w, dw_b, u3);
    }
    k_gn_stats<<<B_, 256, 0, stream>>>(u3, mean2, rstd2, CO * T2);
    k_gn2_apply<<<(B_ * CO * T2 + 255) / 256, 256, 0, stream>>>(
        u3, mean2, rstd2, gn2_g, gn2_b, u4);
    // pw2: u5[b] = pw2_w(32x32) * u4[b](32x2019) + b
    {
        dim3 g((T2 + 63) / 64, CO / 16, B_);
        wmma_gemm_f32<<<g, 32, 0, stream>>>(pw2_w, 0, CO, 1,
                                            u4, (long long)CO * T2, T2, 1,
                                            u5, (long long)CO * T2, T2, 1,
                                            pw2_b, 1, CO, T2, CO);
    }
    // encoder: h[b] = enc_w^T(64x32) * u5[b] + enc_b  (A transposed via strides)
    {
        dim3 g((T2 + 63) / 64, H_ / 16, B_);
        wmma_gemm_f32<<<g, 32, 0, stream>>>(enc_w, 0, 1, H_,
                                            u5, (long long)CO * T2, T2, 1,
                                            hb, (long long)H_ * T2, T2, 1,
                                            enc_b, 1, H_, T2, CO);
    }
    // 8 S4D blocks
    for (int l = 0; l < NL; ++l) {
        k_scan<<<B_ * H_, 32, 0, stream>>>(hb,
            wre + l * H_ * NS, wim + l * H_ * NS,
            cre + l * H_ * NS, cim + l * H_ * NS,
            s4_D + l * H_, yb);
        dim3 g((T2 + 63) / 64, (2 * H_) / 16, B_);
        wmma_gemm_f32<<<g, 32, 0, stream>>>(s4_ow + (long long)l * 2 * H_ * H_, 0, H_, 1,
                                            yb, (long long)H_ * T2, T2, 1,
                                            ob, (long long)2 * H_ * T2, T2, 1,
                                            s4_ob + l * 2 * H_, 1, 2 * H_, T2, H_);
        k_glu_res_ln<<<(B_ * T2 + 255) / 256, 256, 0, stream>>>(
            ob, hb, ln_g + l * H_, ln_b + l * H_);
    }
    // decoder: out[b](2019x640) = h[b]^T(2019x64) * dec_w(64x640) + dec_b(col)
    {
        dim3 g((640 + 63) / 64, (T2 + 15) / 16, B_);
        wmma_gemm_f32<<<g, 32, 0, stream>>>(hb, (long long)H_ * T2, 1, T2,
                                            dec_w, 0, 640, 1,
                                            out, (long long)T2 * 640, 640, 1,
                                            dec_b, 2, T2, 640, 64);
    }
}